// GAT_37812892074107
// MI455X (gfx1250) — compile-verified
//
#include <hip/hip_runtime.h>
#include <math.h>

typedef __attribute__((ext_vector_type(2))) float v2f;
typedef __attribute__((ext_vector_type(8))) float v8f;

#define NEG_SLOPE 0.2f

// Order-preserving uint encoding of float (for deterministic atomic max)
__device__ __forceinline__ unsigned fenc(float x) {
  unsigned u = __float_as_uint(x);
  return (u & 0x80000000u) ? ~u : (u | 0x80000000u);
}
__device__ __forceinline__ float fdec(unsigned e) {
  unsigned b = (e & 0x80000000u) ? (e & 0x7FFFFFFFu) : ~e;
  return __uint_as_float(b);
}

// ---------------------------------------------------------------------------
// GEMM1: H[N,128] = X[N,128] @ W[128,128], fp32 WMMA 16x16x4.
// Block = 256 threads = 8 waves; wave w owns column tile w (16 cols).
// Each block covers 16 rows. N = 50000 = 3125 * 16 exactly.
// A 16x4 layout : lanes 0-15 -> M=lane, K={0,1}; lanes 16-31 -> K={2,3}.
// B 4x16 layout : lanes 0-15 -> N=lane, K={0,1}; lanes 16-31 -> K={2,3}.
// C/D 16x16     : v[i], lane<16 -> row=i, col=lane; lane>=16 -> row=i+8.
// ---------------------------------------------------------------------------
__global__ __launch_bounds__(256) void gat_gemm_128x128(
    const float* __restrict__ X, const float* __restrict__ W,
    float* __restrict__ H)
{
  const int lane = threadIdx.x & 31;
  const int wave = threadIdx.x >> 5;          // 0..7 -> col tile
  const int row0 = blockIdx.x << 4;
  const int col0 = wave << 4;
  const int aRow = row0 + (lane & 15);
  const int kOff = (lane >> 4) << 1;          // 0 or 2
  const int bCol = col0 + (lane & 15);
  const float* xr = X + (size_t)aRow * 128;
  v8f acc = {};
  for (int k = 0; k < 128; k += 4) {
    v2f a, b;
    a.x = xr[k + kOff];
    a.y = xr[k + kOff + 1];
    b.x = W[(size_t)(k + kOff) * 128 + bCol];
    b.y = W[(size_t)(k + kOff + 1) * 128 + bCol];
    acc = __builtin_amdgcn_wmma_f32_16x16x4_f32(
        false, a, false, b, (short)0, acc, false, false);
  }
  const int r = row0 + ((lane >> 4) << 3);
  const int c = col0 + (lane & 15);
#pragma unroll
  for (int i = 0; i < 8; ++i)
    H[(size_t)(r + i) * 128 + c] = acc[i];
}

// ---------------------------------------------------------------------------
// GEMM2: G[N,40] = X[N,128] @ W[128,40]. 3 waves (96 threads), cols masked.
// ---------------------------------------------------------------------------
__global__ __launch_bounds__(96) void gat_gemm_128x40(
    const float* __restrict__ X, const float* __restrict__ W,
    float* __restrict__ G)
{
  const int lane = threadIdx.x & 31;
  const int wave = threadIdx.x >> 5;          // 0..2 -> col tile
  const int row0 = blockIdx.x << 4;
  const int col0 = wave << 4;
  const int aRow = row0 + (lane & 15);
  const int kOff = (lane >> 4) << 1;
  const int bCol = col0 + (lane & 15);
  const bool colOk = (bCol < 40);
  const float* xr = X + (size_t)aRow * 128;
  v8f acc = {};
  for (int k = 0; k < 128; k += 4) {
    v2f a, b;
    a.x = xr[k + kOff];
    a.y = xr[k + kOff + 1];
    b.x = colOk ? W[(size_t)(k + kOff) * 40 + bCol] : 0.0f;
    b.y = colOk ? W[(size_t)(k + kOff + 1) * 40 + bCol] : 0.0f;
    acc = __builtin_amdgcn_wmma_f32_16x16x4_f32(
        false, a, false, b, (short)0, acc, false, false);
  }
  const int r = row0 + ((lane >> 4) << 3);
  const int c = col0 + (lane & 15);
  if (c < 40) {
#pragma unroll
    for (int i = 0; i < 8; ++i)
      G[(size_t)(r + i) * 40 + c] = acc[i];
  }
}

// ---------------------------------------------------------------------------
// Per-node attention dot products: al_src[n,h], al_dst[n,h]
// ---------------------------------------------------------------------------
__global__ __launch_bounds__(256) void gat_attn(
    const float* __restrict__ Hf, const float* __restrict__ aS,
    const float* __restrict__ aD, float* __restrict__ alS,
    float* __restrict__ alD, int N, int NH, int C)
{
  int idx = blockIdx.x * blockDim.x + threadIdx.x;
  if (idx >= N * NH) return;
  int n = idx / NH, h = idx - n * NH;
  const float* hp = Hf + (size_t)n * NH * C + (size_t)h * C;
  const float* sp = aS + (size_t)h * C;
  const float* dp = aD + (size_t)h * C;
  float s = 0.f, d = 0.f;
  for (int c = 0; c < C; ++c) { s += hp[c] * sp[c]; d += hp[c] * dp[c]; }
  alS[idx] = s;
  alD[idx] = d;
}

__device__ __forceinline__ void edge_ends(const int* ei, int E, int e,
                                          int& s, int& d) {
  if (e < E) { s = ei[e]; d = ei[E + e]; }
  else       { s = d = e - E; }               // self loop
}

// Segment max over destinations (order-invariant encoded atomicMax).
__global__ __launch_bounds__(256) void gat_edge_max(
    const int* __restrict__ ei, int E, int NH,
    const float* __restrict__ alS, const float* __restrict__ alD,
    unsigned* __restrict__ mx, int total)
{
  int idx = blockIdx.x * blockDim.x + threadIdx.x;
  if (idx >= total) return;
  int e = idx / NH, h = idx - e * NH;
  int s, d; edge_ends(ei, E, e, s, d);
  float v = alS[s * NH + h] + alD[d * NH + h];
  v = (v > 0.f) ? v : NEG_SLOPE * v;          // leaky relu
  atomicMax(&mx[d * NH + h], fenc(v));
}

// exp(logit - max) per edge, accumulate denominators.
__global__ __launch_bounds__(256) void gat_edge_exp(
    const int* __restrict__ ei, int E, int NH,
    const float* __restrict__ alS, const float* __restrict__ alD,
    const unsigned* __restrict__ mx, float* __restrict__ wexp,
    float* __restrict__ den, int total)
{
  int idx = blockIdx.x * blockDim.x + threadIdx.x;
  if (idx >= total) return;
  int e = idx / NH, h = idx - e * NH;
  int s, d; edge_ends(ei, E, e, s, d);
  float v = alS[s * NH + h] + alD[d * NH + h];
  v = (v > 0.f) ? v : NEG_SLOPE * v;
  float w = expf(v - fdec(mx[d * NH + h]));
  wexp[idx] = w;
  atomicAdd(&den[d * NH + h], w);
}

// Weighted scatter, vectorized: one thread per (edge, 4-channel group).
// out[dst, f..f+3] += w[e,h] * H[src, f..f+3]. C % 4 == 0 and head
// boundaries are multiples of 4, so h is uniform within a group.
__global__ __launch_bounds__(256) void gat_edge_scatter4(
    const int* __restrict__ ei, int E, int NH, int C,
    const float* __restrict__ Hf, const float* __restrict__ wexp,
    float* __restrict__ outA, long long total)
{
  long long idx = (long long)blockIdx.x * blockDim.x + threadIdx.x;
  if (idx >= total) return;
  const int F = NH * C;
  const int Q = F >> 2;                 // float4 groups per edge
  int e = (int)(idx / Q);
  int q = (int)(idx - (long long)e * Q);
  int f = q << 2;
  int s, d; edge_ends(ei, E, e, s, d);
  int h = f / C;
  float w = wexp[(size_t)e * NH + h];
  const float4 hv = *(const float4*)(Hf + (size_t)s * F + f);
  float* op = outA + (size_t)d * F + f;
  atomicAdd(op + 0, w * hv.x);
  atomicAdd(op + 1, w * hv.y);
  atomicAdd(op + 2, w * hv.z);
  atomicAdd(op + 3, w * hv.w);
}

// Layer-1 finalize (in place): h2 = elu(out/(den+eps) + b1)
__global__ __launch_bounds__(256) void gat_finalize1(
    float* __restrict__ out1, const float* __restrict__ den,
    const float* __restrict__ b, int N, int F, int C)
{
  int idx = blockIdx.x * blockDim.x + threadIdx.x;
  if (idx >= N * F) return;
  int n = idx / F, f = idx - n * F;
  int h = f / C;
  float v = out1[idx] / (den[n * (F / C) + h] + 1e-16f) + b[f];
  out1[idx] = (v > 0.f) ? v : (expf(v) - 1.0f);  // ELU
}

// Layer-2 finalize: out = out2/(den+eps) + b2  (1 head -> mean == identity)
__global__ __launch_bounds__(256) void gat_finalize2(
    const float* __restrict__ out2, const float* __restrict__ den,
    const float* __restrict__ b, float* __restrict__ out, int N, int C)
{
  int idx = blockIdx.x * blockDim.x + threadIdx.x;
  if (idx >= N * C) return;
  int n = idx / C, c = idx - n * C;
  out[idx] = out2[idx] / (den[n] + 1e-16f) + b[c];
}

extern "C" void kernel_launch(void* const* d_in, const int* in_sizes, int n_in,
                              void* d_out, int out_size, void* d_ws, size_t ws_size,
                              hipStream_t stream)
{
  const float* x   = (const float*)d_in[0];
  const int*   ei  = (const int*)d_in[1];
  const float* W1  = (const float*)d_in[2];
  const float* aS1 = (const float*)d_in[3];
  const float* aD1 = (const float*)d_in[4];
  const float* b1  = (const float*)d_in[5];
  const float* W2  = (const float*)d_in[6];
  const float* aS2 = (const float*)d_in[7];
  const float* aD2 = (const float*)d_in[8];
  const float* b2  = (const float*)d_in[9];
  float* out = (float*)d_out;

  const int N  = in_sizes[0] / 128;   // 50000
  const int E  = in_sizes[1] / 2;     // 800000
  const int EE = E + N;               // with self loops

  // ---- workspace layout (reused across layers) ----
  char* ws = (char*)d_ws;
  size_t off = 0;
  auto carve = [&](size_t bytes) -> char* {
    char* p = ws + off;
    off += (bytes + 255) & ~(size_t)255;
    return p;
  };
  float*    h1   = (float*)carve((size_t)N * 128 * 4);   // layer1 features
  float*    out1 = (float*)carve((size_t)N * 128 * 4);   // layer1 accum -> h2
  float*    alS  = (float*)carve((size_t)N * 2 * 4);
  float*    alD  = (float*)carve((size_t)N * 2 * 4);
  unsigned* mx   = (unsigned*)carve((size_t)N * 2 * 4);
  float*    den  = (float*)carve((size_t)N * 2 * 4);
  float*    wexp = (float*)carve((size_t)EE * 2 * 4);
  // layer 2 reuses the h1 region: g2 in [0, N*40), out2 in [N*64, N*104)
  float* g2   = h1;
  float* out2 = h1 + (size_t)N * 64;

  // ---------------- Layer 1 ----------------
  gat_gemm_128x128<<<N / 16, 256, 0, stream>>>(x, W1, h1);
  gat_attn<<<(N * 2 + 255) / 256, 256, 0, stream>>>(h1, aS1, aD1, alS, alD, N, 2, 64);
  hipMemsetAsync(mx,   0, (size_t)N * 2 * 4, stream);
  hipMemsetAsync(den,  0, (size_t)N * 2 * 4, stream);
  hipMemsetAsync(out1, 0, (size_t)N * 128 * 4, stream);
  gat_edge_max<<<(EE * 2 + 255) / 256, 256, 0, stream>>>(ei, E, 2, alS, alD, mx, EE * 2);
  gat_edge_exp<<<(EE * 2 + 255) / 256, 256, 0, stream>>>(ei, E, 2, alS, alD, mx, wexp, den, EE * 2);
  {
    long long tot = (long long)EE * 32;   // (e, float4 group) pairs, F=128
    gat_edge_scatter4<<<(unsigned)((tot + 255) / 256), 256, 0, stream>>>(
        ei, E, 2, 64, h1, wexp, out1, tot);
  }
  gat_finalize1<<<(N * 128 + 255) / 256, 256, 0, stream>>>(out1, den, b1, N, 128, 64);

  // ---------------- Layer 2 ----------------
  gat_gemm_128x40<<<N / 16, 96, 0, stream>>>(out1, W2, g2);
  gat_attn<<<(N + 255) / 256, 256, 0, stream>>>(g2, aS2, aD2, alS, alD, N, 1, 40);
  hipMemsetAsync(mx,   0, (size_t)N * 4, stream);
  hipMemsetAsync(den,  0, (size_t)N * 4, stream);
  hipMemsetAsync(out2, 0, (size_t)N * 40 * 4, stream);
  gat_edge_max<<<(EE + 255) / 256, 256, 0, stream>>>(ei, E, 1, alS, alD, mx, EE);
  gat_edge_exp<<<(EE + 255) / 256, 256, 0, stream>>>(ei, E, 1, alS, alD, mx, wexp, den, EE);
  {
    long long tot = (long long)EE * 10;   // F=40 -> 10 float4 groups per edge
    gat_edge_scatter4<<<(unsigned)((tot + 255) / 256), 256, 0, stream>>>(
        ei, E, 1, 40, g2, wexp, out2, tot);
  }
  gat_finalize2<<<(N * 40 + 255) / 256, 256, 0, stream>>>(out2, den, b2, out, N, 40);
}